// GearNet_56324201120047
// MI455X (gfx1250) — compile-verified
//
#include <hip/hip_runtime.h>

typedef __attribute__((ext_vector_type(2))) float v2f;
typedef __attribute__((ext_vector_type(8))) float v8f;

#define N_NODES 50000
#define N_EDGES 600000
#define DIM     128
#define NREL    7
#define NGRAPH  32
#define RD      (NREL * DIM)        /* 896 */
#define NPAD    50048               /* 391 * 128 */
#define LPITCH  66                  /* LDS pitch (floats): 8B-aligned, spreads banks */

// ---------------------------------------------------------------------------
// Scatter: agg[(node_out*R + rel)*128 + d] += h[node_in*128 + d] * w
// One wave per edge; each lane handles a float4 (32 lanes * 4 = 128 feats).
// ---------------------------------------------------------------------------
__global__ __launch_bounds__(256) void scatter_kernel(
    const float* __restrict__ h, const int* __restrict__ node_in,
    const int* __restrict__ node_out, const int* __restrict__ relation,
    const float* __restrict__ ew, float* __restrict__ agg)
{
    int gid = blockIdx.x * blockDim.x + threadIdx.x;   // E*32 threads exactly
    int e   = gid >> 5;
    int q   = (gid & 31) << 2;
    int src = node_in[e];
    float w = ew[e];
    long long dst = (long long)node_out[e] * NREL + relation[e];
    const float4 v = *(const float4*)(h + (long long)src * DIM + q);
    float* a = agg + dst * DIM + q;
    unsafeAtomicAdd(a + 0, v.x * w);
    unsafeAtomicAdd(a + 1, v.y * w);
    unsafeAtomicAdd(a + 2, v.z * w);
    unsafeAtomicAdd(a + 3, v.w * w);
}

// ---------------------------------------------------------------------------
// Fused GEMM: hout = relu(agg[Npad,896] @ lin_w + h[Npad,128] @ sl_w + b) + h
// 8 waves/block; each wave computes a 16x128 strip with V_WMMA_F32_16X16X4_F32.
// Weight 64x128 chunks staged transposed in LDS ([col][k], pitch 66).
// ---------------------------------------------------------------------------
__global__ __launch_bounds__(256) void gemm_fused_kernel(
    const float* __restrict__ agg,     // [NPAD, 896]
    const float* __restrict__ h,       // [NPAD, 128]
    const float* __restrict__ lin_w,   // [896, 128]
    const float* __restrict__ lin_b,   // [128]
    const float* __restrict__ sl_w,    // [128, 128]
    const float* __restrict__ sl_b,    // [128]
    float* __restrict__ hout)          // [NPAD, 128]
{
    __shared__ float lw[DIM * LPITCH];

    const int lane = threadIdx.x & 31;
    const int wv   = threadIdx.x >> 5;     // wave 0..7
    const int lo   = lane & 15;            // M (A) / N (B,C)
    const int hi   = lane >> 4;            // K-group / M-group select

    const long long m0 = (long long)(blockIdx.x * 8 + wv) * 16;

    v8f acc[8] = {};

    for (int chunk = 0; chunk < 16; ++chunk) {
        const int kc = chunk * 64;
        const float* Wsrc;
        const float* Asrc;
        long long astride;
        if (kc < RD) {
            Wsrc = lin_w + (long long)kc * DIM;
            Asrc = agg + m0 * RD + kc;
            astride = RD;
        } else {
            Wsrc = sl_w + (long long)(kc - RD) * DIM;
            Asrc = h + m0 * DIM + (kc - RD);
            astride = DIM;
        }
        __syncthreads();
        // cooperative transposed stage: lw[col*66 + k] = Wsrc[k*128 + col]
        for (int i = threadIdx.x; i < 64 * DIM; i += 256) {
            int krel = i >> 7;
            int col  = i & 127;
            lw[col * LPITCH + krel] = Wsrc[krel * DIM + col];
        }
        __syncthreads();

        const float* ap = Asrc + (long long)lo * astride + 2 * hi;
        #pragma unroll
        for (int kk = 0; kk < 64; kk += 4) {
            // A 16x4 frag: lane holds A[lo][kk+2*hi +{0,1}]  (one global b64)
            v2f a = *(const v2f*)(ap + kk);
            #pragma unroll
            for (int j = 0; j < 8; ++j) {
                // B 4x16 frag: b[vgpr] = W[kk+2*hi+vgpr][j*16+lo] (one ds b64)
                v2f b = *(const v2f*)(&lw[(j * 16 + lo) * LPITCH + kk + 2 * hi]);
                acc[j] = __builtin_amdgcn_wmma_f32_16x16x4_f32(
                    false, a, false, b, (short)0, acc[j], false, false);
            }
        }
    }

    // Epilogue: bias + relu + residual. C/D layout: VGPR v, lane -> M=v+8*hi, N=lo
    #pragma unroll
    for (int j = 0; j < 8; ++j) {
        const int col   = j * 16 + lo;
        const float bias = lin_b[col] + sl_b[col];
        #pragma unroll
        for (int v = 0; v < 8; ++v) {
            const long long m = m0 + v + 8 * hi;
            float r = acc[j][v] + bias;
            r = r > 0.f ? r : 0.f;
            r += h[m * DIM + col];
            hout[m * DIM + col] = r;
        }
    }
}

// ---------------------------------------------------------------------------
// Readout: graph_feature[g] += h[n] for node2graph[n]==g (atomic segment sum)
// ---------------------------------------------------------------------------
__global__ __launch_bounds__(256) void readout_kernel(
    const float* __restrict__ h, const int* __restrict__ n2g,
    float* __restrict__ gf)
{
    int gid = blockIdx.x * blockDim.x + threadIdx.x;   // N*32 threads exactly
    int n = gid >> 5;
    int q = (gid & 31) << 2;
    int g = n2g[n];
    const float4 v = *(const float4*)(h + (long long)n * DIM + q);
    float* o = gf + g * DIM + q;
    unsafeAtomicAdd(o + 0, v.x);
    unsafeAtomicAdd(o + 1, v.y);
    unsafeAtomicAdd(o + 2, v.z);
    unsafeAtomicAdd(o + 3, v.w);
}

// ---------------------------------------------------------------------------
extern "C" void kernel_launch(void* const* d_in, const int* in_sizes, int n_in,
                              void* d_out, int out_size, void* d_ws, size_t ws_size,
                              hipStream_t stream)
{
    const float* x        = (const float*)d_in[0];
    const int*   node_in  = (const int*)d_in[1];
    const int*   node_out = (const int*)d_in[2];
    const int*   relation = (const int*)d_in[3];
    const float* ew       = (const float*)d_in[4];
    const int*   n2g      = (const int*)d_in[5];
    // d_in[6]=num_node, d_in[7]=num_graphs (scalars, compile-time constants here)
    const float* sl_w[3]  = {(const float*)d_in[8],  (const float*)d_in[12], (const float*)d_in[16]};
    const float* sl_b[3]  = {(const float*)d_in[9],  (const float*)d_in[13], (const float*)d_in[17]};
    const float* lin_w[3] = {(const float*)d_in[10], (const float*)d_in[14], (const float*)d_in[18]};
    const float* lin_b[3] = {(const float*)d_in[11], (const float*)d_in[15], (const float*)d_in[19]};

    float* out = (float*)d_out;   // [G*D] graph_feature, then [N*D] node_feature

    // Workspace layout (bytes):
    //   agg : NPAD*896*4  = 179,372,032
    //   hA  : NPAD*128*4  =  25,624,576
    //   hB  : NPAD*128*4  =  25,624,576
    char*  ws  = (char*)d_ws;
    float* agg = (float*)ws;
    float* hA  = (float*)(ws + (size_t)NPAD * RD * sizeof(float));
    float* hB  = hA + (size_t)NPAD * DIM;

    // h0 = x (pad rows stay garbage; they only feed pad outputs, never read)
    hipMemcpyAsync(hA, x, (size_t)N_NODES * DIM * sizeof(float),
                   hipMemcpyDeviceToDevice, stream);

    float* hin  = hA;
    float* hnext = hB;
    for (int t = 0; t < 3; ++t) {
        hipMemsetAsync(agg, 0, (size_t)NPAD * RD * sizeof(float), stream);
        scatter_kernel<<<(N_EDGES * 32) / 256, 256, 0, stream>>>(
            hin, node_in, node_out, relation, ew, agg);
        gemm_fused_kernel<<<NPAD / 128, 256, 0, stream>>>(
            agg, hin, lin_w[t], lin_b[t], sl_w[t], sl_b[t], hnext);
        float* tmp = hin; hin = hnext; hnext = tmp;
    }

    // graph readout + node feature copy-out
    hipMemsetAsync(out, 0, (size_t)NGRAPH * DIM * sizeof(float), stream);
    readout_kernel<<<(N_NODES * 32) / 256, 256, 0, stream>>>(hin, n2g, out);
    hipMemcpyAsync(out + NGRAPH * DIM, hin, (size_t)N_NODES * DIM * sizeof(float),
                   hipMemcpyDeviceToDevice, stream);
}